// CausalSelfAttention_89532888252903
// MI455X (gfx1250) — compile-verified
//
#include <hip/hip_runtime.h>
#include <math.h>

typedef __attribute__((ext_vector_type(2))) float v2f;
typedef __attribute__((ext_vector_type(8))) float v8f;
typedef unsigned int v4u __attribute__((ext_vector_type(4)));
typedef int v4i __attribute__((ext_vector_type(4)));
typedef int v8i __attribute__((ext_vector_type(8)));

__device__ __forceinline__ v8f wmma_f32(v2f a, v2f b, v8f c) {
  // V_WMMA_F32_16X16X4_F32 : D = A(16x4) x B(4x16) + C(16x16), fp32
  return __builtin_amdgcn_wmma_f32_16x16x4_f32(false, a, false, b, (short)0, c,
                                               false, false);
}

__device__ __forceinline__ unsigned int lds_off(const void* p) {
  // generic-AS LDS pointer: low 32 bits are the LDS byte address
  return (unsigned int)(unsigned long long)p;
}

// ---------------------------------------------------------------------------
// TDM 2D tile load: tile_d1 rows of tile_d0 fp32 elements, row stride
// stride0 elements, into LDS at lds_byte with hardware padding so the LDS
// row stride becomes (tile_d0 + pad) dwords.
//   pad_interval: encoded, interval = 2^(v+1) dwords (must equal tile_d0)
//   pad_amount:   encoded, amount   = v+1 dwords
// ---------------------------------------------------------------------------
__device__ __forceinline__ void tdm_load_2d(unsigned int lds_byte,
                                            const float* gptr,
                                            unsigned int tile_d0,
                                            unsigned int tile_d1,
                                            unsigned long long stride0,
                                            unsigned int pad_interval,
                                            unsigned int pad_amount) {
  unsigned long long ga = (unsigned long long)gptr;
  v4u g0;
  g0[0] = 1u;                                   // count=1, user descriptor
  g0[1] = lds_byte;                             // lds_addr
  g0[2] = (unsigned int)ga;                     // global_addr[31:0]
  g0[3] = (unsigned int)((ga >> 32) & 0x1FFFFFFu) | (2u << 30); // addr[56:32], type=2

  v8i g1;
  g1[0] = (int)((2u << 16) |                    // data_size = 4 bytes
                (1u << 20) |                    // pad_enable
                (pad_interval << 22) |
                (pad_amount << 25));
  g1[1] = (int)((tile_d0 & 0xFFFFu) << 16);     // tensor_dim0[15:0] (== tile)
  g1[2] = (int)((tile_d0 >> 16) | ((tile_d1 & 0xFFFFu) << 16)); // d0 hi | tensor_dim1 lo
  g1[3] = (int)((tile_d1 >> 16) | (tile_d0 << 16));             // d1 hi | tile_dim0
  g1[4] = (int)(tile_d1 & 0xFFFFu);             // tile_dim1 (tile_dim2 = 0)
  g1[5] = (int)(unsigned int)(stride0 & 0xFFFFFFFFu);           // tensor_dim0_stride lo
  g1[6] = (int)(unsigned int)(stride0 >> 32);                   // stride hi (dim1_stride=0)
  g1[7] = 0;

  v4i z4 = {0, 0, 0, 0};
#if defined(__clang_major__) && __clang_major__ >= 23
  v8i z8 = {0, 0, 0, 0, 0, 0, 0, 0};
  __builtin_amdgcn_tensor_load_to_lds(g0, g1, z4, z4, z8, 0);
#else
  __builtin_amdgcn_tensor_load_to_lds(g0, g1, z4, z4, 0);
#endif
}

// ---------------------------------------------------------------------------
// GEMM: C[M,N] = A[M,K] @ B[K,N] + bias[N]   (M%128==0, N%128==0, K%16==0)
// 256 threads = 8 waves; block tile 128x128; wave tile 32x64; K-tile 16.
// TDM double-buffered LDS staging; fp32 WMMA 16x16x4 compute.
// ---------------------------------------------------------------------------
__global__ __launch_bounds__(256) void gemm_bias_wmma(
    const float* __restrict__ A, const float* __restrict__ Bm,
    const float* __restrict__ bias, float* __restrict__ Cm,
    int Mdim, int Ndim, int Kdim)
{
  __shared__ float Alds[2][128][17];   // TDM pad: 16 dwords + 1  (interval v=3, amount v=0)
  __shared__ float Blds[2][16][132];   // TDM pad: 128 dwords + 4 (interval v=6, amount v=3)

  const int tid  = threadIdx.x;
  const int lane = tid & 31;
  const int wid  = tid >> 5;
  const int l15  = lane & 15;
  const int hi2  = (lane >> 4) << 1;     // 0 (lanes 0-15) or 2 (lanes 16-31)
  const int hi8  = (lane >> 4) << 3;     // 0 or 8 (C/D row offset)
  const int wm   = (wid & 3) * 32;
  const int wn   = (wid >> 2) * 64;
  const int tm   = blockIdx.y * 128;
  const int tn   = blockIdx.x * 128;

  v8f acc[2][4];
  #pragma unroll
  for (int mi = 0; mi < 2; ++mi)
    #pragma unroll
    for (int ni = 0; ni < 4; ++ni)
      #pragma unroll
      for (int e = 0; e < 8; ++e) acc[mi][ni][e] = 0.0f;

  const int nkt = Kdim >> 4;

  if (wid == 0) {   // wave-uniform: wave 0 drives the TDM pipeline
    tdm_load_2d(lds_off(&Alds[0][0][0]), A + (size_t)tm * Kdim,
                16u, 128u, (unsigned long long)Kdim, 3u, 0u);
    tdm_load_2d(lds_off(&Blds[0][0][0]), Bm + tn,
                128u, 16u, (unsigned long long)Ndim, 6u, 3u);
  }

  for (int kt = 0; kt < nkt; ++kt) {
    const int cur = kt & 1;
    if (wid == 0) __builtin_amdgcn_s_wait_tensorcnt(0);  // tile kt resident
    __syncthreads();                                     // publish to all waves
    if (wid == 0 && (kt + 1) < nkt) {                    // prefetch tile kt+1
      const int k0n = (kt + 1) << 4;
      tdm_load_2d(lds_off(&Alds[1 - cur][0][0]),
                  A + (size_t)tm * Kdim + k0n,
                  16u, 128u, (unsigned long long)Kdim, 3u, 0u);
      tdm_load_2d(lds_off(&Blds[1 - cur][0][0]),
                  Bm + (size_t)k0n * Ndim + tn,
                  128u, 16u, (unsigned long long)Ndim, 6u, 3u);
    }

    #pragma unroll
    for (int kk = 0; kk < 4; ++kk) {
      const int ka = kk * 4 + hi2;                 // K index pair for this half-wave
      v2f a[2], b[4];
      #pragma unroll
      for (int mi = 0; mi < 2; ++mi) {
        int r = wm + mi * 16 + l15;                // A frag: M = lane, K = ka,ka+1
        a[mi][0] = Alds[cur][r][ka];
        a[mi][1] = Alds[cur][r][ka + 1];
      }
      #pragma unroll
      for (int ni = 0; ni < 4; ++ni) {
        int c = wn + ni * 16 + l15;                // B frag: N = lane, K = ka,ka+1
        b[ni][0] = Blds[cur][ka][c];
        b[ni][1] = Blds[cur][ka + 1][c];
      }
      #pragma unroll
      for (int mi = 0; mi < 2; ++mi)
        #pragma unroll
        for (int ni = 0; ni < 4; ++ni)
          acc[mi][ni] = wmma_f32(a[mi], b[ni], acc[mi][ni]);
    }
    __syncthreads();   // compute done before this buffer is overwritten (kt+2)
  }

  // epilogue: D layout — vgpr e holds (M = e + hi8, N = l15)
  #pragma unroll
  for (int ni = 0; ni < 4; ++ni) {
    int col = tn + wn + ni * 16 + l15;
    float bv = bias[col];
    #pragma unroll
    for (int mi = 0; mi < 2; ++mi) {
      #pragma unroll
      for (int e = 0; e < 8; ++e) {
        int row = tm + wm + mi * 16 + e + hi8;
        Cm[(size_t)row * Ndim + col] = acc[mi][ni][e] + bv;
      }
    }
  }
}

// ---------------------------------------------------------------------------
// Flash attention over qkv[B*T, 3C] (q|k|v per row), head-strided access.
// Block = (b, h, 64 q-rows); 4 waves, wave w owns q rows [qb+16w, qb+16w+16).
// K/V 16x64 tiles staged by TDM (double-buffered); online softmax wave-local.
// ---------------------------------------------------------------------------
__global__ __launch_bounds__(128) void attn_flash_wmma(
    const float* __restrict__ qkv, float* __restrict__ y,
    int Bsz, int T, int C, int NH, int HS)
{
  __shared__ float Klds[2][16][68];   // TDM pad: 64 dwords + 4 (interval v=5, amount v=3)
  __shared__ float Vlds[2][16][68];
  __shared__ float Plds[4][16][17];   // per-wave D->A layout staging

  const int tid  = threadIdx.x;
  const int lane = tid & 31;
  const int w    = tid >> 5;          // wave 0..3
  const int l15  = lane & 15;
  const int hi2  = (lane >> 4) << 1;
  const int hi8  = (lane >> 4) << 3;

  const int qb = blockIdx.x * 64;
  const int h  = blockIdx.y;
  const int b  = blockIdx.z;

  const size_t rs = (size_t)3 * C;                      // qkv row stride
  const float* qbase = qkv + (size_t)b * T * rs + h * HS;
  const float* kbase = qbase + C;
  const float* vbase = qbase + 2 * C;

  // Preload this wave's Q fragments (16 rows x 64 d) in A layout.
  const int qrow = qb + 16 * w + l15;
  const float* qptr = qbase + (size_t)qrow * rs;
  float qa0[16], qa1[16];
  #pragma unroll
  for (int kk = 0; kk < 16; ++kk) {
    int d = kk * 4 + hi2;
    qa0[kk] = qptr[d];
    qa1[kk] = qptr[d + 1];
  }

  v8f o[4];
  #pragma unroll
  for (int nb = 0; nb < 4; ++nb)
    #pragma unroll
    for (int e = 0; e < 8; ++e) o[nb][e] = 0.0f;

  float mrow[8], lrow[8];
  #pragma unroll
  for (int e = 0; e < 8; ++e) { mrow[e] = -INFINITY; lrow[e] = 0.0f; }

  const int myqmax = qb + 16 * w + 15;
  const int ntiles = (qb + 64) >> 4;  // key tiles 0..ntiles-1 (causal upper bound)
  const float scale = 0.125f;         // 1/sqrt(64)

  if (w == 0) {
    tdm_load_2d(lds_off(&Klds[0][0][0]), kbase, 64u, 16u,
                (unsigned long long)rs, 5u, 3u);
    tdm_load_2d(lds_off(&Vlds[0][0][0]), vbase, 64u, 16u,
                (unsigned long long)rs, 5u, 3u);
  }

  for (int kt = 0; kt < ntiles; ++kt) {
    const int cur = kt & 1;
    if (w == 0) __builtin_amdgcn_s_wait_tensorcnt(0);
    __syncthreads();
    if (w == 0 && (kt + 1) < ntiles) {
      const float* ksrc = kbase + (size_t)((kt + 1) * 16) * rs;
      const float* vsrc = vbase + (size_t)((kt + 1) * 16) * rs;
      tdm_load_2d(lds_off(&Klds[1 - cur][0][0]), ksrc, 64u, 16u,
                  (unsigned long long)rs, 5u, 3u);
      tdm_load_2d(lds_off(&Vlds[1 - cur][0][0]), vsrc, 64u, 16u,
                  (unsigned long long)rs, 5u, 3u);
    }

    if (kt * 16 <= myqmax) {          // wave-uniform: tile visible to this wave
      // S = Q @ K^T  (contract d = 64 -> 16 WMMA steps)
      v8f s;
      #pragma unroll
      for (int e = 0; e < 8; ++e) s[e] = 0.0f;
      #pragma unroll
      for (int kk = 0; kk < 16; ++kk) {
        int dk = kk * 4 + hi2;
        v2f a, bf;
        a[0] = qa0[kk]; a[1] = qa1[kk];
        bf[0] = Klds[cur][l15][dk];   // B[d][n] = K[n][d]
        bf[1] = Klds[cur][l15][dk + 1];
        s = wmma_f32(a, bf, s);
      }

      // Online softmax update (rows live in 16-lane groups).
      const int col = kt * 16 + l15;
      #pragma unroll
      for (int e = 0; e < 8; ++e) {
        int row = qb + 16 * w + e + hi8;
        float sv = s[e] * scale;
        if (col > row) sv = -INFINITY;
        float rmax = sv;
        rmax = fmaxf(rmax, __shfl_xor(rmax, 1));
        rmax = fmaxf(rmax, __shfl_xor(rmax, 2));
        rmax = fmaxf(rmax, __shfl_xor(rmax, 4));
        rmax = fmaxf(rmax, __shfl_xor(rmax, 8));
        float mnew = fmaxf(mrow[e], rmax);
        float p = __expf(sv - mnew);
        float rsum = p;
        rsum += __shfl_xor(rsum, 1);
        rsum += __shfl_xor(rsum, 2);
        rsum += __shfl_xor(rsum, 4);
        rsum += __shfl_xor(rsum, 8);
        float corr = __expf(mrow[e] - mnew);
        lrow[e] = lrow[e] * corr + rsum;
        mrow[e] = mnew;
        #pragma unroll
        for (int nb = 0; nb < 4; ++nb) o[nb][e] *= corr;
        Plds[w][e + hi8][l15] = p;    // stage P (D layout -> memory)
      }

      // O += P @ V  (contract keys = 16 -> 4 WMMA steps, 4 d-blocks)
      #pragma unroll
      for (int kk = 0; kk < 4; ++kk) {
        int pk = kk * 4 + hi2;
        v2f a;
        a[0] = Plds[w][l15][pk];      // A layout: M = lane, K = pk,pk+1
        a[1] = Plds[w][l15][pk + 1];
        #pragma unroll
        for (int nb = 0; nb < 4; ++nb) {
          v2f bf;
          int c = nb * 16 + l15;
          bf[0] = Vlds[cur][pk][c];
          bf[1] = Vlds[cur][pk + 1][c];
          o[nb] = wmma_f32(a, bf, o[nb]);
        }
      }
    }
    __syncthreads();
  }

  // Normalize and write y[B,T,C] at column h*HS + d.
  float inv[8];
  #pragma unroll
  for (int e = 0; e < 8; ++e) inv[e] = 1.0f / lrow[e];
  #pragma unroll
  for (int nb = 0; nb < 4; ++nb) {
    int col = h * HS + nb * 16 + l15;
    #pragma unroll
    for (int e = 0; e < 8; ++e) {
      int row = qb + 16 * w + e + hi8;
      y[((size_t)b * T + row) * C + col] = o[nb][e] * inv[e];
    }
  }
}

// ---------------------------------------------------------------------------
extern "C" void kernel_launch(void* const* d_in, const int* in_sizes, int n_in,
                              void* d_out, int out_size, void* d_ws, size_t ws_size,
                              hipStream_t stream) {
  (void)in_sizes; (void)n_in; (void)out_size; (void)ws_size;
  const float* x      = (const float*)d_in[0];
  const float* W_attn = (const float*)d_in[1];
  const float* b_attn = (const float*)d_in[2];
  const float* W_proj = (const float*)d_in[3];
  const float* b_proj = (const float*)d_in[4];
  float* out = (float*)d_out;

  const int B = 4, T = 2048, C = 1024, NH = 16, HS = 64;
  const int M = B * T;                       // 8192 rows

  float* qkv = (float*)d_ws;                 // [M, 3C]  = 96 MB
  float* y   = qkv + (size_t)M * 3 * C;      // [M, C]   = 32 MB

  // 1) qkv = x @ W_attn + b_attn
  gemm_bias_wmma<<<dim3(3 * C / 128, M / 128), dim3(256), 0, stream>>>(
      x, W_attn, b_attn, qkv, M, 3 * C, C);

  // 2) y = causal-softmax(q k^T / sqrt(HS)) v   (flash, per head)
  attn_flash_wmma<<<dim3(T / 64, NH, B), dim3(128), 0, stream>>>(
      qkv, y, B, T, C, NH, HS);

  // 3) out = y @ W_proj + b_proj
  gemm_bias_wmma<<<dim3(C / 128, M / 128), dim3(256), 0, stream>>>(
      y, W_proj, b_proj, out, M, C, C);
}